// PaiNNMixing_34394098106847
// MI455X (gfx1250) — compile-verified
//
#include <hip/hip_runtime.h>
#include <hip/hip_bf16.h>

typedef __attribute__((ext_vector_type(2))) float v2f;
typedef __attribute__((ext_vector_type(8))) float v8f;

#define FD      512
#define TM      16
#define THREADS 256
#define EPS_F   1e-8f

// LDS layout (float offsets), total 73728 floats = 288 KB
#define OFF_MUW   0        // mu_W  [3][16][512]
#define OFF_MUVX  24576    // mu_V  [3][16][512]  (reused later as x [16][1536])
#define OFF_VN    49152    // mu_Vn [16][512]
#define OFF_VV    57344    // sum_d mu_V*mu_W [16][512]
#define OFF_H     65536    // h     [16][512]
#define LDS_FLOATS 73728

static __device__ __forceinline__ v8f wmma4(v2f a, v2f b, v8f c) {
  // V_WMMA_F32_16X16X4_F32: full-precision fp32 matrix op (D = A(16x4) * B(4x16) + C)
  return __builtin_amdgcn_wmma_f32_16x16x4_f32(false, a, false, b, (short)0, c,
                                               false, false);
}

__global__ __launch_bounds__(THREADS) void painn_mixing_kernel(
    const float* __restrict__ q, const float* __restrict__ mu,
    const float* __restrict__ Wmix, const float* __restrict__ W1,
    const float* __restrict__ b1, const float* __restrict__ W2,
    const float* __restrict__ b2,
    float* __restrict__ q_out, float* __restrict__ mu_out, int N)
{
  extern __shared__ float smem[];
  float* s_muW = smem + OFF_MUW;
  float* s_muV = smem + OFF_MUVX;
  float* s_vn  = smem + OFF_VN;
  float* s_vv  = smem + OFF_VV;
  float* s_h   = smem + OFF_H;
  float* s_x   = smem + OFF_MUVX;   // reuse after mu_V is consumed

  const int tid   = threadIdx.x;
  const int wave  = tid >> 5;       // 8 wave32 per block
  const int lane  = tid & 31;
  const int n16   = lane & 15;      // M row (A) / N col (B) within 16x16 tile
  const int half  = lane >> 4;      // selects K pair {0,1} vs {2,3}
  const int sbase = blockIdx.x * TM;

  int srow = sbase + n16;
  if (srow >= N) srow = N - 1;      // clamp reads; writes are guarded below

  const v8f vzero = {};

  // ---------------- Stage A: mu_mix = mu(16x512, per d) @ Wmix^T(512x1024) ----
  {
    const int gw = wave * 128;                       // 8 tiles of 16 columns
    float* dst = (gw < FD) ? s_muV : s_muW;          // waves 0-3 -> V, 4-7 -> W
    const int gofs = (gw < FD) ? gw : (gw - FD);
    for (int d = 0; d < 3; ++d) {
      v8f acc[8];
#pragma unroll
      for (int t = 0; t < 8; ++t) acc[t] = vzero;
      const float* arow  = mu + (size_t)srow * (3 * FD) + d * FD + 2 * half;
      const float* bbase = Wmix + 2 * half;
      for (int k = 0; k < FD; k += 4) {
        v2f a = *(const v2f*)(arow + k);
#pragma unroll
        for (int t = 0; t < 8; ++t) {
          const int g = gw + t * 16 + n16;
          v2f b = *(const v2f*)(bbase + (size_t)g * FD + k);
          acc[t] = wmma4(a, b, acc[t]);
        }
      }
#pragma unroll
      for (int t = 0; t < 8; ++t) {
#pragma unroll
        for (int r = 0; r < 8; ++r) {
          const int m = r + 8 * half;                // C/D layout: VGPR r, half
          const int g = gofs + t * 16 + n16;
          dst[(d * TM + m) * FD + g] = acc[t][r];
        }
      }
    }
  }
  __syncthreads();

  // ---------------- Stage A2: vn = sqrt(sum_d V^2 + eps), vv = sum_d V*W -----
  for (int i = tid; i < TM * FD; i += THREADS) {
    const int s = i >> 9, f = i & (FD - 1);
    const float v0 = s_muV[(0 * TM + s) * FD + f];
    const float v1 = s_muV[(1 * TM + s) * FD + f];
    const float v2 = s_muV[(2 * TM + s) * FD + f];
    const float w0 = s_muW[(0 * TM + s) * FD + f];
    const float w1 = s_muW[(1 * TM + s) * FD + f];
    const float w2 = s_muW[(2 * TM + s) * FD + f];
    s_vn[i] = sqrtf(v0 * v0 + v1 * v1 + v2 * v2 + EPS_F);
    s_vv[i] = v0 * w0 + v1 * w1 + v2 * w2;
  }
  __syncthreads();

  // ---------------- Stage B: h = silu([q, vn](16x1024) @ W1^T + b1) ----------
  {
    const int gw = wave * 64;                        // 4 tiles of 16 columns
    v8f acc[4];
#pragma unroll
    for (int t = 0; t < 4; ++t) acc[t] = vzero;

    // K = 0..511 : ctx = q  (global, L2-hot)
    const float* arow = q + (size_t)srow * FD + 2 * half;
    for (int k = 0; k < FD; k += 4) {
      v2f a = *(const v2f*)(arow + k);
#pragma unroll
      for (int t = 0; t < 4; ++t) {
        const int g = gw + t * 16 + n16;
        v2f b = *(const v2f*)(W1 + (size_t)g * (2 * FD) + k + 2 * half);
        acc[t] = wmma4(a, b, acc[t]);
      }
    }
    // K = 512..1023 : ctx = vn (LDS)
    const float* vrow = s_vn + n16 * FD + 2 * half;
    for (int k = 0; k < FD; k += 4) {
      v2f a = *(const v2f*)(vrow + k);
#pragma unroll
      for (int t = 0; t < 4; ++t) {
        const int g = gw + t * 16 + n16;
        v2f b = *(const v2f*)(W1 + (size_t)g * (2 * FD) + FD + k + 2 * half);
        acc[t] = wmma4(a, b, acc[t]);
      }
    }
#pragma unroll
    for (int t = 0; t < 4; ++t) {
#pragma unroll
      for (int r = 0; r < 8; ++r) {
        const int m = r + 8 * half;
        const int g = gw + t * 16 + n16;
        const float v = acc[t][r] + b1[g];
        s_h[m * FD + g] = v / (1.0f + __expf(-v));   // silu
      }
    }
  }
  __syncthreads();

  // ---------------- Stage C: x = h(16x512) @ W2^T(512x1536) + b2 -------------
  {
    const int gw = wave * 192;                       // 12 tiles of 16 columns
    v8f acc[12];
#pragma unroll
    for (int t = 0; t < 12; ++t) acc[t] = vzero;
    const float* arow = s_h + n16 * FD + 2 * half;
    for (int k = 0; k < FD; k += 4) {
      v2f a = *(const v2f*)(arow + k);
#pragma unroll
      for (int t = 0; t < 12; ++t) {
        const int g = gw + t * 16 + n16;
        v2f b = *(const v2f*)(W2 + (size_t)g * FD + k + 2 * half);
        acc[t] = wmma4(a, b, acc[t]);
      }
    }
#pragma unroll
    for (int t = 0; t < 12; ++t) {
#pragma unroll
      for (int r = 0; r < 8; ++r) {
        const int m = r + 8 * half;
        const int g = gw + t * 16 + n16;
        s_x[m * (3 * FD) + g] = acc[t][r] + b2[g];   // mu_V buffer is dead now
      }
    }
  }
  __syncthreads();

  // ---------------- Stage D: outputs -----------------------------------------
  for (int i = tid; i < TM * FD; i += THREADS) {
    const int s = i >> 9, f = i & (FD - 1);
    const int gs = sbase + s;
    if (gs < N) {
      const size_t off = (size_t)gs * FD + f;
      const float x0 = s_x[s * (3 * FD) + f];
      const float x2 = s_x[s * (3 * FD) + 2 * FD + f];
      q_out[off] = q[off] + x0 + x2 * s_vv[s * FD + f];
    }
  }
  for (int s = 0; s < TM; ++s) {
    const int gs = sbase + s;
    if (gs >= N) break;
#pragma unroll
    for (int d = 0; d < 3; ++d) {
      for (int f = tid; f < FD; f += THREADS) {
        const size_t off = (size_t)gs * (3 * FD) + d * FD + f;
        const float x1 = s_x[s * (3 * FD) + FD + f];
        mu_out[off] = mu[off] + x1 * s_muW[(d * TM + s) * FD + f];
      }
    }
  }
}

extern "C" void kernel_launch(void* const* d_in, const int* in_sizes, int n_in,
                              void* d_out, int out_size, void* d_ws, size_t ws_size,
                              hipStream_t stream) {
  const float* q    = (const float*)d_in[0];
  const float* mu   = (const float*)d_in[1];
  const float* Wmix = (const float*)d_in[2];
  const float* W1   = (const float*)d_in[3];
  const float* b1   = (const float*)d_in[4];
  const float* W2   = (const float*)d_in[5];
  const float* b2   = (const float*)d_in[6];

  const int N = in_sizes[0] / FD;                    // q is (N,1,F)
  float* q_out  = (float*)d_out;                     // (N,1,F) then (N,3,F)
  float* mu_out = q_out + (size_t)N * FD;

  const int blocks = (N + TM - 1) / TM;
  const size_t shmem = (size_t)LDS_FLOATS * sizeof(float);  // 288 KB (<320 KB/WGP)

  (void)hipFuncSetAttribute(reinterpret_cast<const void*>(painn_mixing_kernel),
                            hipFuncAttributeMaxDynamicSharedMemorySize,
                            (int)shmem);

  if (blocks > 0) {
    painn_mixing_kernel<<<blocks, THREADS, shmem, stream>>>(
        q, mu, Wmix, W1, b1, W2, b2, q_out, mu_out, N);
  }
}